// EPGSimulationMQC_89618787598336
// MI455X (gfx1250) — compile-verified
//
#include <hip/hip_runtime.h>
#include <math.h>

// EPG-MQC recurrence for MI455X (gfx1250, wave32).
// One wave = one batch element; lane <-> EPG state via a ROTATING map:
//   after pulse k, lane L holds state s = (L + k) & 31.
// This makes the per-pulse EPG shift free for F (zero one lane, relabel) and
// reduces cross-lane traffic to a single ds_bpermute per pulse (Z, which does
// NOT shift in state space, must move -1 lane to stay with its state).
// Pulse tables staged in LDS via gfx1250 async DMA (global_load_async_to_lds_b128).
// Output: 369 MB write-once -> non-temporal streaming stores (>> 192 MB L2).

#define N_PULSES 128
#define BATCH    2048
#define N_STATES 32
#define WAVES_PER_BLOCK 8

typedef float f2 __attribute__((ext_vector_type(2)));

__global__ __launch_bounds__(WAVES_PER_BLOCK * 32)
void epg_mqc_kernel(const float* __restrict__ fap,   // (128) flip angles
                    const float* __restrict__ php,   // (128) pulse phases
                    const float* __restrict__ T1p,   // (2048)
                    const float* __restrict__ T2p,
                    const float* __restrict__ B0p,
                    const float* __restrict__ B1p,
                    const int*   __restrict__ TRp,   // scalar
                    float* __restrict__ outF,        // (128,2048,32,5) complex64 (re/im)
                    float* __restrict__ outZ)        // (128,2048,32) f32
{
    __shared__ __align__(16) float lds_fa[N_PULSES];
    __shared__ __align__(16) float lds_ph[N_PULSES];

    const int lane = threadIdx.x & 31;
    const int wave = threadIdx.x >> 5;
    const int b    = blockIdx.x * WAVES_PER_BLOCK + wave;

    // ---- stage the 2x512B pulse tables into LDS with async DMA (ASYNCcnt) ----
    if (wave == 0) {
        unsigned fa_lds = (unsigned)(uintptr_t)(&lds_fa[0]) + (unsigned)lane * 16u;
        unsigned ph_lds = (unsigned)(uintptr_t)(&lds_ph[0]) + (unsigned)lane * 16u;
        const float* fa_g = fap + lane * 4;   // 16 B per lane, 32 lanes = 512 B
        const float* ph_g = php + lane * 4;
        asm volatile("global_load_async_to_lds_b128 %0, %2, off\n\t"
                     "global_load_async_to_lds_b128 %1, %3, off\n\t"
                     "s_wait_asynccnt 0x0"
                     :
                     : "v"(fa_lds), "v"(ph_lds), "v"(fa_g), "v"(ph_g)
                     : "memory");
    }
    __syncthreads();

    // ---- per-batch (wave-uniform) constants ----
    const float TRf = (float)TRp[0];
    const float E1  = __expf(-TRf / T1p[b]);
    const float E2  = __expf(-TRf / T2p[b]);
    const float rec = 1.0f - E1;
    const float B1  = B1p[b];
    const float phi0 = 6.28318530717958647692f * B0p[b] * TRf * 0.001f;
    float s1, c1;
    __sincosf(phi0, &s1, &c1);               // e^{i*phi0}  (fast HW trig)
    const float c2 = c1 * c1 - s1 * s1;      // e^{i*2*phi0}
    const float s2 = 2.0f * c1 * s1;

    // Z rotates -1 lane per pulse: lane L reads from lane (L+1)&31
    const int zsrc = ((lane + 1) & 31) << 2;

    // ---- state registers (labels rotate; data mostly stationary) ----
    float m2r = 0.f, m2i = 0.f;   // q = -2
    float m1r = 0.f, m1i = 0.f;   // q = -1
    float p1r = 0.f, p1i = 0.f;   // q = +1
    float p2r = 0.f, p2i = 0.f;   // q = +2
    float Z   = (lane == 0) ? 1.0f : 0.0f;   // identity map at n=0

    for (int n = 0; n < N_PULSES; ++n) {
        const float fa = lds_fa[n];          // ds_load_b32, wave-uniform
        const float ph = lds_ph[n];

        // lane currently labeled state 0 (gets T1 recovery)
        const int lrec  = (-n) & 31;
        // lane whose NEW label (post-shift) is state 0 (F components zeroed)
        const int lzero = (31 - n) & 31;
        // state label after this pulse (used for store addressing)
        const int s     = (lane + n + 1) & 31;

        // ---- relaxation ----
        m2r *= E2; m2i *= E2; m1r *= E2; m1i *= E2;
        p1r *= E2; p1i *= E2; p2r *= E2; p2i *= E2;
        Z *= E1;
        if (lane == lrec) Z += rec;

        // ---- B0 phase accrual: F_q *= e^{i*q*phi0} ----
        float t;
        t = m2r * c2 + m2i * s2;  m2i = m2i * c2 - m2r * s2;  m2r = t;  // q=-2
        t = m1r * c1 + m1i * s1;  m1i = m1i * c1 - m1r * s1;  m1r = t;  // q=-1
        t = p1r * c1 - p1i * s1;  p1i = p1i * c1 + p1r * s1;  p1r = t;  // q=+1
        t = p2r * c2 - p2i * s2;  p2i = p2i * c2 + p2r * s2;  p2r = t;  // q=+2

        // ---- RF mixing (B1-scaled flip, phase ph) ----
        float sa, ca;
        __sincosf(0.5f * fa * B1, &sa, &ca);
        const float ca2 = ca * ca, sa2 = sa * sa, casa = ca * sa;
        float ei, er;
        __sincosf(ph, &ei, &er);                                  // eib = er + i*ei
        const float e2r = er * er - ei * ei,   e2i = 2.f * er * ei;    // eib^2
        const float e4r = e2r * e2r - e2i * e2i, e4i = 2.f * e2r * e2i; // eib^4

        const float czei = casa * Z * ei;
        const float czer = casa * Z * er;

        // Fp1' = ca^2*Fp1 + sa^2*conj(Fm1)*eib^2 + i*ca*sa*Z*eib
        const float np1r = ca2 * p1r + sa2 * (m1r * e2r + m1i * e2i) - czei;
        const float np1i = ca2 * p1i + sa2 * (m1r * e2i - m1i * e2r) + czer;
        // Fm1' = sa^2*conj(Fp1)*conj(eib^2) + ca^2*Fm1 - i*ca*sa*Z*conj(eib)
        const float nm1r = ca2 * m1r + sa2 * (p1r * e2r - p1i * e2i) - czei;
        const float nm1i = ca2 * m1i - sa2 * (p1r * e2i + p1i * e2r) - czer;
        // Z' = Re(-i*ca*sa*(Fm1*eib - Fp1*conj(eib))) + (ca^2-sa^2)*Z
        const float nZ = casa * (ei * (m1r + p1r) + er * (m1i - p1i)) + (ca2 - sa2) * Z;
        // Fp2' = ca^2*Fp2 + sa^2*conj(Fm2)*eib^4
        const float np2r = ca2 * p2r + sa2 * (m2r * e4r + m2i * e4i);
        const float np2i = ca2 * p2i + sa2 * (m2r * e4i - m2i * e4r);
        // Fm2' = sa^2*conj(Fp2)*conj(eib^4) + ca^2*Fm2
        const float nm2r = ca2 * m2r + sa2 * (p2r * e4r - p2i * e4i);
        const float nm2i = ca2 * m2i - sa2 * (p2r * e4i + p2i * e4r);

        // ---- EPG shift via relabeling: zero the lane that becomes state 0 ----
        const bool kz = (lane == lzero);
        m2r = kz ? 0.f : nm2r;  m2i = kz ? 0.f : nm2i;
        m1r = kz ? 0.f : nm1r;  m1i = kz ? 0.f : nm1i;
        p1r = kz ? 0.f : np1r;  p1i = kz ? 0.f : np1i;
        p2r = kz ? 0.f : np2r;  p2i = kz ? 0.f : np2i;

        // ---- Z does NOT shift in state space -> move -1 lane to track the map ----
        Z = __int_as_float(__builtin_amdgcn_ds_bpermute(zsrc, __float_as_int(nZ)));

        // ---- streaming non-temporal stores ----
        // F_all[n,b,s,m]: 10 floats per (n,b,s); channel order [-2,-1,0,+1,+2].
        // Lanes cover a rotated-but-complete contiguous 1280 B tile per wave.
        const size_t row = (size_t)n * BATCH + b;
        float* fptr = outF + row * (N_STATES * 10) + (size_t)s * 10;
        f2 v0 = {m2r, m2i};
        f2 v1 = {m1r, m1i};
        f2 v2 = {0.0f, 0.0f};                  // q=0 channel is identically zero
        f2 v3 = {p1r, p1i};
        f2 v4 = {p2r, p2i};
        __builtin_nontemporal_store(v0, (f2*)(fptr + 0));
        __builtin_nontemporal_store(v1, (f2*)(fptr + 2));
        __builtin_nontemporal_store(v2, (f2*)(fptr + 4));
        __builtin_nontemporal_store(v3, (f2*)(fptr + 6));
        __builtin_nontemporal_store(v4, (f2*)(fptr + 8));
        __builtin_nontemporal_store(Z, outZ + row * N_STATES + s);
    }
}

extern "C" void kernel_launch(void* const* d_in, const int* in_sizes, int n_in,
                              void* d_out, int out_size, void* d_ws, size_t ws_size,
                              hipStream_t stream) {
    const float* fap = (const float*)d_in[0];   // flip_angles (128)
    const float* php = (const float*)d_in[1];   // phases      (128)
    const float* T1p = (const float*)d_in[2];   // (2048)
    const float* T2p = (const float*)d_in[3];
    const float* B0p = (const float*)d_in[4];
    const float* B1p = (const float*)d_in[5];
    const int*   TRp = (const int*)d_in[6];     // scalar TR
    // d_in[7] = TE (unused by the reference core)

    float* outF = (float*)d_out;
    float* outZ = outF + (size_t)N_PULSES * BATCH * N_STATES * 5 * 2;

    dim3 block(WAVES_PER_BLOCK * 32);
    dim3 grid(BATCH / WAVES_PER_BLOCK);
    epg_mqc_kernel<<<grid, block, 0, stream>>>(fap, php, T1p, T2p, B0p, B1p, TRp, outF, outZ);
}